// overall_25228637897611
// MI455X (gfx1250) — compile-verified
//
#include <hip/hip_runtime.h>
#include <hip/hip_bf16.h>

typedef float wv2f __attribute__((ext_vector_type(2)));
typedef float wv8f __attribute__((ext_vector_type(8)));

#define EPSV 1e-5f
#define SQRT1_2F 0.70710678118654752f

__device__ __forceinline__ float gelu_exact(float v) {
    return 0.5f * v * (1.0f + erff(v * SQRT1_2F));
}

// ---------------------------------------------------------------------------
// Kernel 1: adaptive avg pool 96x96 -> 3x3 window means. One block per (b,c).
// x1s[bc*9 + (ky*3+kx)] = mean of 32x32 window.
// ---------------------------------------------------------------------------
__global__ __launch_bounds__(256) void pool_kernel(const float* __restrict__ x,
                                                   float* __restrict__ x1s) {
    const int bc  = blockIdx.x;      // 0..2047
    const int tid = threadIdx.x;     // 0..255
    __shared__ float red[9 * 256];
    const float4* xp = reinterpret_cast<const float4*>(x + (size_t)bc * 9216);

#pragma unroll
    for (int w = 0; w < 9; ++w) red[w * 256 + tid] = 0.0f;
    // each thread only touches red[w*256+tid] below; no barrier needed yet

#pragma unroll
    for (int k = 0; k < 9; ++k) {
        int i = tid + 256 * k;              // float4 index 0..2303
        float4 f = xp[i];
        int y  = i / 24;                    // row 0..95 (24 float4 per row)
        int xq = i - y * 24;                // float4 col 0..23
        int w  = (y >> 5) * 3 + (xq >> 3);  // 32-wide windows (float4-aligned)
        red[w * 256 + tid] += f.x + f.y + f.z + f.w;
    }
    __syncthreads();
    if (tid < 9) {
        float s = 0.0f;
        for (int j = 0; j < 256; ++j) s += red[tid * 256 + j];
        x1s[bc * 9 + tid] = s * (1.0f / 1024.0f);
    }
}

// ---------------------------------------------------------------------------
// Kernel 2: H1 = gelu(bn(c0_w[64,128] @ X[128,144] + c0_b)) via WMMA f32 16x16x4
// X[i][col] = x1s[(b*128+i)*9+kl], col = b*9+kl, N = 16*9 = 144 = 9 tiles exactly.
// One wave (32 threads) per 16x16 output tile. grid = (4 mtiles, 9 ntiles).
// ---------------------------------------------------------------------------
__global__ __launch_bounds__(32) void gemm1_kernel(
    const float* __restrict__ c0_w, const float* __restrict__ c0_b,
    const float* __restrict__ bn0_g, const float* __restrict__ bn0_b,
    const float* __restrict__ bn0_m, const float* __restrict__ bn0_v,
    const float* __restrict__ x1s, float* __restrict__ H1) {
    const int mtile = blockIdx.x;           // 0..3
    const int ntile = blockIdx.y;           // 0..8
    const int lane  = threadIdx.x;          // 0..31
    const int m16   = lane & 15;
    const int hi    = lane >> 4;
    const int col   = ntile * 16 + m16;     // 0..143
    const int bb    = col / 9;
    const int kl    = col - bb * 9;
    const int arow  = mtile * 16 + m16;     // A row (out channel)

    wv8f acc = {0.f, 0.f, 0.f, 0.f, 0.f, 0.f, 0.f, 0.f};
    for (int k0 = 0; k0 < 128; k0 += 4) {
        const int ka = k0 + 2 * hi;
        wv2f a, b;
        a[0] = c0_w[arow * 128 + ka];
        a[1] = c0_w[arow * 128 + ka + 1];
        b[0] = x1s[(bb * 128 + ka) * 9 + kl];
        b[1] = x1s[(bb * 128 + ka + 1) * 9 + kl];
        acc = __builtin_amdgcn_wmma_f32_16x16x4_f32(
            false, a, false, b, (short)0, acc, false, false);
    }
#pragma unroll
    for (int v = 0; v < 8; ++v) {
        const int row = mtile * 16 + v + 8 * hi;   // 0..63
        float val = acc[v] + c0_b[row];
        val = (val - bn0_m[row]) * (rsqrtf(bn0_v[row] + EPSV) * bn0_g[row]) + bn0_b[row];
        H1[row * 144 + col] = gelu_exact(val);
    }
}

// ---------------------------------------------------------------------------
// Kernel 3: scale1 = c1_w[128,64] @ H1[64,144] + c1_b ; 2-way softmax against
// x1s ; blend w_dyn[0],w_dyn[1] -> dynamic depthwise weights wout[B*C,9].
// grid = (8 mtiles, 9 ntiles), 1 wave each.
// ---------------------------------------------------------------------------
__global__ __launch_bounds__(32) void gemm2_kernel(
    const float* __restrict__ c1_w, const float* __restrict__ c1_b,
    const float* __restrict__ H1, const float* __restrict__ x1s,
    const float* __restrict__ w_dyn, float* __restrict__ wout) {
    const int mtile = blockIdx.x;           // 0..7
    const int ntile = blockIdx.y;           // 0..8
    const int lane  = threadIdx.x;
    const int m16   = lane & 15;
    const int hi    = lane >> 4;
    const int col   = ntile * 16 + m16;     // 0..143
    const int bb    = col / 9;
    const int kl    = col - bb * 9;
    const int arow  = mtile * 16 + m16;     // channel for A rows

    wv8f acc = {0.f, 0.f, 0.f, 0.f, 0.f, 0.f, 0.f, 0.f};
    for (int k0 = 0; k0 < 64; k0 += 4) {
        const int ka = k0 + 2 * hi;
        wv2f a, b;
        a[0] = c1_w[arow * 64 + ka];
        a[1] = c1_w[arow * 64 + ka + 1];
        b[0] = H1[ka * 144 + col];
        b[1] = H1[(ka + 1) * 144 + col];
        acc = __builtin_amdgcn_wmma_f32_16x16x4_f32(
            false, a, false, b, (short)0, acc, false, false);
    }
#pragma unroll
    for (int v = 0; v < 8; ++v) {
        const int c = mtile * 16 + v + 8 * hi;    // out channel 0..127
        float s1 = acc[v] + c1_b[c];
        float xv = x1s[(bb * 128 + c) * 9 + kl];
        float mx = fmaxf(s1, xv);
        float e0 = expf(s1 - mx);
        float e1 = expf(xv - mx);
        float inv = 1.0f / (e0 + e1);
        float wv = (e0 * w_dyn[c * 9 + kl] + e1 * w_dyn[(128 + c) * 9 + kl]) * inv;
        wout[(bb * 128 + c) * 9 + kl] = wv;
    }
}

// ---------------------------------------------------------------------------
// Kernel 4: dynamic bias MLP per batch (tiny). 1 block per b, 128 threads.
// ---------------------------------------------------------------------------
__global__ __launch_bounds__(128) void biasmlp_kernel(
    const float* __restrict__ x1s,
    const float* __restrict__ d0_w, const float* __restrict__ d0_b,
    const float* __restrict__ bnd_g, const float* __restrict__ bnd_b,
    const float* __restrict__ bnd_m, const float* __restrict__ bnd_v,
    const float* __restrict__ d1_w, const float* __restrict__ d1_b,
    const float* __restrict__ b_dyn, float* __restrict__ bout) {
    const int b = blockIdx.x;
    const int t = threadIdx.x;   // 0..127
    __shared__ float gap[128];
    __shared__ float h[64];

    {   // gap = plane mean = mean of the 9 uniform window means
        float s = 0.0f;
#pragma unroll
        for (int kl = 0; kl < 9; ++kl) s += x1s[(b * 128 + t) * 9 + kl];
        gap[t] = s * (1.0f / 9.0f);
    }
    __syncthreads();
    if (t < 64) {
        float s = d0_b[t];
        for (int i = 0; i < 128; ++i) s += d0_w[t * 128 + i] * gap[i];
        s = (s - bnd_m[t]) * (rsqrtf(bnd_v[t] + EPSV) * bnd_g[t]) + bnd_b[t];
        h[t] = gelu_exact(s);
    }
    __syncthreads();
    {
        float s0 = d1_b[t];
        float s1 = d1_b[128 + t];
        for (int i = 0; i < 64; ++i) {
            s0 += d1_w[t * 64 + i] * h[i];
            s1 += d1_w[(128 + t) * 64 + i] * h[i];
        }
        float mx = fmaxf(s0, s1);
        float e0 = expf(s0 - mx);
        float e1 = expf(s1 - mx);
        float inv = 1.0f / (e0 + e1);
        bout[b * 128 + t] = (e0 * b_dyn[t] + e1 * b_dyn[128 + t]) * inv;
    }
}

// ---------------------------------------------------------------------------
// Kernel 5: per-plane depthwise 3x3 conv (pad 1) + bias. Bandwidth-critical:
// one block per (b,c) plane; zero-padded LDS tile; register sliding window.
// Block (96,4): thread = output column x, 24 output rows each.
// ---------------------------------------------------------------------------
__global__ __launch_bounds__(384) void dwconv_kernel(
    const float* __restrict__ x, const float* __restrict__ wdyn,
    const float* __restrict__ bias, float* __restrict__ out) {
    const int bc  = blockIdx.x;                 // 0..2047
    const int tx  = threadIdx.x;                // 0..95 (output column)
    const int ty  = threadIdx.y;                // 0..3
    const int tid = ty * 96 + tx;               // 0..383
    const int S   = 100;                        // padded LDS row stride (floats)
    __shared__ float sm[98 * 100];              // (96+2) x (96+2), stride 100

    for (int i = tid; i < 98 * 100; i += 384) sm[i] = 0.0f;
    __syncthreads();

    const float4* xp = reinterpret_cast<const float4*>(x + (size_t)bc * 9216);
    for (int i = tid; i < 2304; i += 384) {
        float4 f = xp[i];
        int y  = i / 24;
        int xq = i - y * 24;
        int base = (y + 1) * S + 1 + xq * 4;
        sm[base]     = f.x;
        sm[base + 1] = f.y;
        sm[base + 2] = f.z;
        sm[base + 3] = f.w;
    }

    const float w0 = wdyn[bc * 9 + 0], w1 = wdyn[bc * 9 + 1], w2 = wdyn[bc * 9 + 2];
    const float w3 = wdyn[bc * 9 + 3], w4 = wdyn[bc * 9 + 4], w5 = wdyn[bc * 9 + 5];
    const float w6 = wdyn[bc * 9 + 6], w7 = wdyn[bc * 9 + 7], w8 = wdyn[bc * 9 + 8];
    const float bs = bias[bc];
    __syncthreads();

    // output (y, tx) reads padded rows y..y+2, padded cols tx..tx+2
    const int ybase = ty * 24;
    float a0 = sm[ybase * S + tx], a1 = sm[ybase * S + tx + 1], a2 = sm[ybase * S + tx + 2];
    float b0 = sm[(ybase + 1) * S + tx], b1 = sm[(ybase + 1) * S + tx + 1],
          b2 = sm[(ybase + 1) * S + tx + 2];
    float* op = out + (size_t)bc * 9216;
#pragma unroll 4
    for (int yy = 0; yy < 24; ++yy) {
        const int y = ybase + yy;
        float c0 = sm[(y + 2) * S + tx];
        float c1 = sm[(y + 2) * S + tx + 1];
        float c2 = sm[(y + 2) * S + tx + 2];
        float s = bs;
        s = fmaf(w0, a0, s); s = fmaf(w1, a1, s); s = fmaf(w2, a2, s);
        s = fmaf(w3, b0, s); s = fmaf(w4, b1, s); s = fmaf(w5, b2, s);
        s = fmaf(w6, c0, s); s = fmaf(w7, c1, s); s = fmaf(w8, c2, s);
        op[y * 96 + tx] = s;
        a0 = b0; a1 = b1; a2 = b2;
        b0 = c0; b1 = c1; b2 = c2;
    }
}

// ---------------------------------------------------------------------------
extern "C" void kernel_launch(void* const* d_in, const int* in_sizes, int n_in,
                              void* d_out, int out_size, void* d_ws, size_t ws_size,
                              hipStream_t stream) {
    const float* x     = (const float*)d_in[0];
    const float* w_dyn = (const float*)d_in[1];
    const float* b_dyn = (const float*)d_in[2];
    const float* c0_w  = (const float*)d_in[3];
    const float* c0_b  = (const float*)d_in[4];
    const float* bn0_g = (const float*)d_in[5];
    const float* bn0_b = (const float*)d_in[6];
    const float* bn0_m = (const float*)d_in[7];
    const float* bn0_v = (const float*)d_in[8];
    const float* c1_w  = (const float*)d_in[9];
    const float* c1_b  = (const float*)d_in[10];
    const float* d0_w  = (const float*)d_in[11];
    const float* d0_b  = (const float*)d_in[12];
    const float* bnd_g = (const float*)d_in[13];
    const float* bnd_b = (const float*)d_in[14];
    const float* bnd_m = (const float*)d_in[15];
    const float* bnd_v = (const float*)d_in[16];
    const float* d1_w  = (const float*)d_in[17];
    const float* d1_b  = (const float*)d_in[18];
    float* out = (float*)d_out;

    float* ws   = (float*)d_ws;
    float* x1s  = ws;                       // B*C*9  = 18432 floats
    float* H1   = x1s + 18432;              // 64*144 =  9216 floats
    float* wout = H1 + 9216;                // B*C*9  = 18432 floats
    float* bout = wout + 18432;             // B*C    =  2048 floats

    pool_kernel<<<2048, 256, 0, stream>>>(x, x1s);
    gemm1_kernel<<<dim3(4, 9), 32, 0, stream>>>(c0_w, c0_b, bn0_g, bn0_b, bn0_m,
                                                bn0_v, x1s, H1);
    gemm2_kernel<<<dim3(8, 9), 32, 0, stream>>>(c1_w, c1_b, H1, x1s, w_dyn, wout);
    biasmlp_kernel<<<16, 128, 0, stream>>>(x1s, d0_w, d0_b, bnd_g, bnd_b, bnd_m,
                                           bnd_v, d1_w, d1_b, b_dyn, bout);
    dwconv_kernel<<<2048, dim3(96, 4), 0, stream>>>(x, wout, bout, out);
}